// QtrDecoder_40501541601484
// MI455X (gfx1250) — compile-verified
//
#include <hip/hip_runtime.h>
#include <cstdint>

// ---------------- problem constants (from reference) ----------------
#define D_LAT   128
#define NNODE   512          // N
#define ROWF    132          // floats per node row (D_LAT + 4)
#define HH      128
#define WW      128
#define PPTS    4096         // P points per (b,t)
#define BT      64           // B*T
#define SPLIT   4            // blocks per (b,t)
#define PTS_PER_BLK (PPTS / SPLIT)   // 1024
#define BLK     256          // 8 wave32
#define PTS_PER_THR (PTS_PER_BLK / BLK)  // 4
#define NODE_F  48           // compacted floats per node in LDS
#define LDS_BYTES (NNODE * NODE_F * 4)   // 98304 B = 96 KB < 320 KB WGP LDS

// LDS node layout (floats, stride 48):
//   [0..41]  = coeff columns 0..41
//   [44..47] = source columns 128..131 -> hw0, hw1, (unused), valid_flag
// All async transfers 8/16B aligned (row stride 528 B = 33*16).

typedef int v4i __attribute__((ext_vector_type(4)));
typedef int v2i __attribute__((ext_vector_type(2)));
typedef __attribute__((address_space(1))) v4i* g4p;   // global v4i*
typedef __attribute__((address_space(3))) v4i* l4p;   // LDS v4i*
typedef __attribute__((address_space(1))) v2i* g2p;
typedef __attribute__((address_space(3))) v2i* l2p;

#define ASYNC_B128(g, l, off) \
    __builtin_amdgcn_global_load_async_to_lds_b128((g4p)(g), (l4p)(l), (off), 0)
#define ASYNC_B64(g, l, off) \
    __builtin_amdgcn_global_load_async_to_lds_b64((g2p)(g), (l2p)(l), (off), 0)

__device__ __forceinline__ void wait_async_zero() {
#if __has_builtin(__builtin_amdgcn_s_wait_asynccnt)
    __builtin_amdgcn_s_wait_asynccnt(0);
#else
    asm volatile("s_wait_asynccnt 0x0" ::: "memory");
#endif
}

extern "C" __global__ __launch_bounds__(BLK)
void qtr_decoder_kernel(const float* __restrict__ nodes,   // (BT, N, 132)
                        const int*   __restrict__ seg,     // (BT, H, W)
                        const int*   __restrict__ sp,      // (BT, P, 2)
                        float*       __restrict__ out)     // (BT, P, 7)
{
    extern __shared__ float smem[];   // NNODE * NODE_F floats

    const int bt    = blockIdx.x / SPLIT;
    const int chunk = blockIdx.x % SPLIT;
    const int tid   = threadIdx.x;

    const float* nodes_b = nodes + (size_t)bt * NNODE * ROWF;
    const int*   seg_b   = seg   + (size_t)bt * HH * WW;
    const int*   sp_b    = sp    + (size_t)bt * PPTS * 2;
    float*       out_b   = out   + (size_t)bt * PPTS * 7;

    // ---- stage: compact 512 x 48-float node table into LDS via async DMA.
    // HW adds INST_OFFSET to BOTH the LDS and global addresses (ISA §15.18
    // async pseudocode), so one base address pair per node + literal
    // immediate offsets covers 11/12 transfers; the tail quad (global +512 /
    // LDS +176) pre-biases the global pointer by 336 B (still 16B aligned).
    for (int n = tid; n < NNODE; n += BLK) {
        const float* g = nodes_b + (size_t)n * ROWF;   // row base: 528 B, 16B aligned
        float*       l = smem + n * NODE_F;
        ASYNC_B128(g, l,   0);    // coeff floats  0..3
        ASYNC_B128(g, l,  16);    // coeff floats  4..7
        ASYNC_B128(g, l,  32);    // coeff floats  8..11
        ASYNC_B128(g, l,  48);    // coeff floats 12..15
        ASYNC_B128(g, l,  64);    // coeff floats 16..19
        ASYNC_B128(g, l,  80);    // coeff floats 20..23
        ASYNC_B128(g, l,  96);    // coeff floats 24..27
        ASYNC_B128(g, l, 112);    // coeff floats 28..31
        ASYNC_B128(g, l, 128);    // coeff floats 32..35
        ASYNC_B128(g, l, 144);    // coeff floats 36..39
        ASYNC_B64 (g, l, 160);    // coeff floats 40..41
        ASYNC_B128(g + 84, l, 176); // source cols 128..131 -> LDS floats 44..47
    }

    // ---- overlap: point-index gathers are LDS-independent; issue before wait ----
    const int base = chunk * PTS_PER_BLK;
    int sy[PTS_PER_THR], sx[PTS_PER_THR], sg[PTS_PER_THR];
#pragma unroll
    for (int k = 0; k < PTS_PER_THR; ++k) {
        const int p = base + tid + k * BLK;
        sy[k] = sp_b[2 * p + 0];
        sx[k] = sp_b[2 * p + 1];
    }
#pragma unroll
    for (int k = 0; k < PTS_PER_THR; ++k)
        sg[k] = seg_b[sy[k] * WW + sx[k]];

    wait_async_zero();
    __syncthreads();

    // ---- compute: 4 points per thread, coalesced stride-256 ----
#pragma unroll
    for (int k = 0; k < PTS_PER_THR; ++k) {
        const int p = base + tid + k * BLK;
        const int s = sg[k];

        const float vseg = (s >= 0 && s < NNODE) ? 1.0f : 0.0f;
        int sc = s < 0 ? 0 : (s > (NNODE - 1) ? (NNODE - 1) : s);
        const float* nd = smem + sc * NODE_F;

        const float valid = vseg * (nd[47] > 0.0f ? 1.0f : 0.0f);

        const float sifH = (float)sy[k] / 63.5f - 1.0f;   // scale = (H-1)/2
        const float sifW = (float)sx[k] / 63.5f - 1.0f;
        const float dH = sifH - nd[44];
        const float dW = sifW - nd[45];

        const float del[6] = {1.0f, dH, dW, dH * dH, dH * dW, dW * dW};

        // pred_depths: cols 0..5, min(val, -0.1)
        float depth = 0.0f;
#pragma unroll
        for (int j = 0; j < 6; ++j) depth += nd[j] * del[j];
        depth *= valid;
        depth = fminf(depth, -0.1f);

        // pred_images: cols 6..23, chunk j channel c -> col 6 + 3j + c, clip +-100
        float img[3];
#pragma unroll
        for (int c = 0; c < 3; ++c) {
            float v = 0.0f;
#pragma unroll
            for (int j = 0; j < 6; ++j) v += nd[6 + 3 * j + c] * del[j];
            v *= valid;
            img[c] = fminf(fmaxf(v, -100.0f), 100.0f);
        }

        // pred_normals: cols 24..41, L2-normalized
        float nrm[3];
        float ss = 0.0f;
#pragma unroll
        for (int c = 0; c < 3; ++c) {
            float v = 0.0f;
#pragma unroll
            for (int j = 0; j < 6; ++j) v += nd[24 + 3 * j + c] * del[j];
            v *= valid;
            nrm[c] = v;
            ss += v * v;
        }
        const float inv = rsqrtf(fmaxf(ss, 1e-12f));

        float* o = out_b + (size_t)p * 7;
        o[0] = depth;
        o[1] = img[0]; o[2] = img[1]; o[3] = img[2];
        o[4] = nrm[0] * inv; o[5] = nrm[1] * inv; o[6] = nrm[2] * inv;
    }
}

extern "C" void kernel_launch(void* const* d_in, const int* in_sizes, int n_in,
                              void* d_out, int out_size, void* d_ws, size_t ws_size,
                              hipStream_t stream) {
    (void)in_sizes; (void)n_in; (void)out_size; (void)d_ws; (void)ws_size;
    const float* nodes = (const float*)d_in[0];   // (16,4,512,132) f32
    const int*   seg   = (const int*)d_in[1];     // (16,4,128,128) i32
    const int*   sp    = (const int*)d_in[2];     // (16,4,4096,2) i32
    float*       out   = (float*)d_out;           // (16,4,4096,7) f32

    dim3 grid(BT * SPLIT);   // 256 blocks
    dim3 block(BLK);         // 256 threads = 8 wave32
    qtr_decoder_kernel<<<grid, block, LDS_BYTES, stream>>>(nodes, seg, sp, out);
}